// MaxPool2dAsConv2d_15161234555083
// MI455X (gfx1250) — compile-verified
//
#include <hip/hip_runtime.h>

// ---------------------------------------------------------------------------
// 2x2 valid conv (1->1 ch) + bias + ReLU on (128,1,512,512) f32.
// Memory-bound (AI ~= 1.1 FLOP/B): optimized as an HBM-rate streaming kernel
// using the CDNA5 async-to-LDS data path (GLOBAL_LOAD_ASYNC_TO_LDS_B128,
// tracked by ASYNCcnt) to stage input row tiles, then ds_load_b128 reads.
// WMMA is intentionally not used: K=4 single-channel conv would run the
// matrix pipes at <0.1% utilization and cannot move the HBM roofline.
// ---------------------------------------------------------------------------

#define AS1 __attribute__((address_space(1)))
#define AS3 __attribute__((address_space(3)))

typedef int v4i __attribute__((ext_vector_type(4)));

#if __has_builtin(__builtin_amdgcn_global_load_async_to_lds_b128) && \
    __has_builtin(__builtin_amdgcn_s_wait_asynccnt)
#define USE_ASYNC_LDS 1
#else
#define USE_ASYNC_LDS 0
#endif

constexpr int W     = 512;   // input width
constexpr int H     = 512;   // input height
constexpr int WOUT  = 511;
constexpr int HOUT  = 511;
constexpr int TILEH = 16;    // output rows per block
constexpr int NT    = 256;   // 8 wave32s

__global__ __launch_bounds__(NT)
void conv2x2_bias_relu(const float* __restrict__ x,
                       const float* __restrict__ wgt,
                       const float* __restrict__ bias,
                       float* __restrict__ out)
{
    // 17 rows x 512 floats = 34,816 B of LDS (WGP has 320 KB -> ~9 blocks/WGP)
    __shared__ float smem[(TILEH + 1) * W];

    const int tid  = threadIdx.x;
    const int n    = blockIdx.y;
    const int row0 = blockIdx.x * TILEH;

    const float* xn = x + (size_t)n * H * W;

    // ---- Stage TILEH+1 input rows into LDS via async tensor path ----------
    // Each 16B chunk: 17 rows * 128 chunks = 2176 chunk loads over 256 threads.
    constexpr int CHUNKS = (TILEH + 1) * (W / 4);
    for (int idx = tid; idx < CHUNKS; idx += NT) {
        const int r  = idx >> 7;            // idx / (W/4)
        const int ck = idx & (W / 4 - 1);   // 16B chunk within row
        int gr = row0 + r;
        gr = (gr < H) ? gr : (H - 1);       // clamp at bottom edge (last block)
        const float* gsrc = xn + (size_t)gr * W + ck * 4;
        float*       ldst = &smem[r * W + ck * 4];
#if USE_ASYNC_LDS
        __builtin_amdgcn_global_load_async_to_lds_b128(
            (AS1 v4i*)gsrc, (AS3 v4i*)ldst, /*offset=*/0, /*cpol=*/0);
#else
        *(float4*)ldst = *(const float4*)gsrc;
#endif
    }
#if USE_ASYNC_LDS
    __builtin_amdgcn_s_wait_asynccnt(0);    // ASYNCcnt == 0: LDS tile resident
#endif
    __syncthreads();

    // ---- Per-block scalar params (uniform) --------------------------------
    const float w0 = wgt[0], w1 = wgt[1], w2 = wgt[2], w3 = wgt[3];
    const float bb = bias[0];

    // ---- Compute: each thread owns 4 consecutive output columns -----------
    const int tx = tid & 127;     // column group 0..127
    const int ty = tid >> 7;      // row parity 0..1
    const int c0 = tx * 4;
    int c4 = c0 + 4;
    if (c4 >= W) c4 = W - 1;      // tail clamp (only affects discarded col 511)

    float* outn = out + (size_t)n * HOUT * WOUT;

    for (int rr = ty; rr < TILEH; rr += 2) {
        const int orow = row0 + rr;
        if (orow >= HOUT) break;  // only trims last block

        const float* s0 = &smem[rr * W];
        const float* s1 = &smem[(rr + 1) * W];

        // 16B-aligned LDS vector reads (ds_load_b128) + one tail element
        const float4 A = *(const float4*)&s0[c0];
        const float4 B = *(const float4*)&s1[c0];
        const float  a4 = s0[c4];
        const float  b4 = s1[c4];

        float r0 = fmaf(w0, A.x, fmaf(w1, A.y, fmaf(w2, B.x, fmaf(w3, B.y, bb))));
        float r1 = fmaf(w0, A.y, fmaf(w1, A.z, fmaf(w2, B.y, fmaf(w3, B.z, bb))));
        float r2 = fmaf(w0, A.z, fmaf(w1, A.w, fmaf(w2, B.z, fmaf(w3, B.w, bb))));
        float r3 = fmaf(w0, A.w, fmaf(w1, a4,  fmaf(w2, B.w, fmaf(w3, b4,  bb))));

        r0 = fmaxf(r0, 0.0f);
        r1 = fmaxf(r1, 0.0f);
        r2 = fmaxf(r2, 0.0f);
        r3 = fmaxf(r3, 0.0f);

        // Output row stride 511 floats is not 16B aligned -> scalar b32 stores
        // (lanes still cover a contiguous 512B span per wave: fully coalesced).
        float* op = outn + (size_t)orow * WOUT + c0;
        float  rv[4] = {r0, r1, r2, r3};
#pragma unroll
        for (int k = 0; k < 4; ++k) {
            if (c0 + k < WOUT) op[k] = rv[k];
        }
    }
}

extern "C" void kernel_launch(void* const* d_in, const int* in_sizes, int n_in,
                              void* d_out, int out_size, void* d_ws, size_t ws_size,
                              hipStream_t stream) {
    const float* x   = (const float*)d_in[0];   // (N,1,512,512) f32
    const float* wgt = (const float*)d_in[1];   // (1,1,2,2) f32
    const float* bia = (const float*)d_in[2];   // (1,) f32
    float*       out = (float*)d_out;           // (N,1,511,511) f32

    const int nbatch = in_sizes[0] / (H * W);   // 128
    dim3 grid((HOUT + TILEH - 1) / TILEH, nbatch);
    conv2x2_bias_relu<<<grid, NT, 0, stream>>>(x, wgt, bia, out);
}